// GraphSage_44641890074988
// MI455X (gfx1250) — compile-verified
//
#include <hip/hip_runtime.h>

#define N_NODES 100000
#define N_EDGES 1600000
#define DFEAT 64
#define LDS_PITCH 68  // pad 16x64 f32 tiles to avoid LDS bank conflicts

typedef __attribute__((ext_vector_type(2))) float v2f;
typedef __attribute__((ext_vector_type(8))) float v8f;

// ---------------------------------------------------------------- zero fill
__global__ __launch_bounds__(256) void sage_zero_f32(float* __restrict__ p, int n4) {
    int i = blockIdx.x * blockDim.x + threadIdx.x;
    if (i < n4) {
        float4 z = {0.f, 0.f, 0.f, 0.f};
        ((float4*)p)[i] = z;
    }
}

// ---------------------------------------------------------------- degree count
__global__ __launch_bounds__(256) void sage_count(const long long* __restrict__ dst,
                                                  float* __restrict__ cnt, int nEdges) {
    int e = blockIdx.x * blockDim.x + threadIdx.x;
    if (e < nEdges) {
        long long d = dst[e];
        unsafeAtomicAdd(&cnt[d], 1.0f);  // native global_atomic_add_f32
    }
}

// ---------------------------------------------------------------- scatter-add of features
// One wave32 per edge: lane L moves features L and L+32 (fully coalesced 256B per edge).
__global__ __launch_bounds__(256) void sage_scatter(const long long* __restrict__ src,
                                                    const long long* __restrict__ dst,
                                                    const float* __restrict__ h,
                                                    float* __restrict__ agg, int nEdges) {
    int e = blockIdx.x * 8 + (threadIdx.x >> 5);
    if (e >= nEdges) return;
    int lane = threadIdx.x & 31;
    long long s = src[e];
    long long d = dst[e];
    const float* hs = h + s * (long long)DFEAT;
    float* ad = agg + d * (long long)DFEAT;
    float v0 = hs[lane];
    float v1 = hs[lane + 32];
    unsafeAtomicAdd(&ad[lane], v0);
    unsafeAtomicAdd(&ad[lane + 32], v1);
}

// ---------------------------------------------------------------- fused SAGE layer GEMM
// out[16 rows x 64 cols] per block = mean @ Wl^T + h @ Wr^T + b, optional ReLU.
// 128 threads = 4 waves; wave w owns output cols [16w,16w+16) via V_WMMA_F32_16X16X4_F32.
__global__ __launch_bounds__(128) void sage_gemm(const float* __restrict__ agg,
                                                 const float* __restrict__ cnt,
                                                 const float* __restrict__ h,
                                                 const float* __restrict__ Wl,
                                                 const float* __restrict__ Wr,
                                                 const float* __restrict__ bias,
                                                 float* __restrict__ out, int relu) {
    __shared__ float s_mean[16][LDS_PITCH];
    __shared__ float s_self[16][LDS_PITCH];

    const int row0 = blockIdx.x * 16;  // N_NODES % 16 == 0 -> always full tiles
    const int t = threadIdx.x;

    // Stage mean (agg * 1/max(cnt,1)) and self tiles into LDS: 8 elems per thread each.
    {
        int r = t >> 3;             // 0..15 row in tile
        int c0 = (t & 7) * 8;       // 0,8,...,56
        int row = row0 + r;
        float inv = 1.0f / fmaxf(cnt[row], 1.0f);
        const float* ar = agg + (long long)row * DFEAT + c0;
        const float* hr = h + (long long)row * DFEAT + c0;
#pragma unroll
        for (int i = 0; i < 8; ++i) {
            s_mean[r][c0 + i] = ar[i] * inv;
            s_self[r][c0 + i] = hr[i];
        }
    }
    __syncthreads();

    const int wv = t >> 5;          // wave id 0..3 -> output col tile
    const int lane = t & 31;
    const int m = lane & 15;        // A row / B col within tile
    const int koff = (lane < 16) ? 0 : 2;
    const int ncol = wv * 16 + m;   // global output column == W row

    const float* WlRow = Wl + ncol * DFEAT;
    const float* WrRow = Wr + ncol * DFEAT;

    v8f acc = {0.f, 0.f, 0.f, 0.f, 0.f, 0.f, 0.f, 0.f};

#pragma unroll
    for (int k0 = 0; k0 < DFEAT; k0 += 4) {
        v2f a, b;
        a.x = s_mean[m][k0 + koff];
        a.y = s_mean[m][k0 + koff + 1];
        b.x = WlRow[k0 + koff];
        b.y = WlRow[k0 + koff + 1];
        acc = __builtin_amdgcn_wmma_f32_16x16x4_f32(false, a, false, b, (short)0, acc,
                                                    false, false);
    }
#pragma unroll
    for (int k0 = 0; k0 < DFEAT; k0 += 4) {
        v2f a, b;
        a.x = s_self[m][k0 + koff];
        a.y = s_self[m][k0 + koff + 1];
        b.x = WrRow[k0 + koff];
        b.y = WrRow[k0 + koff + 1];
        acc = __builtin_amdgcn_wmma_f32_16x16x4_f32(false, a, false, b, (short)0, acc,
                                                    false, false);
    }

    // Epilogue: C/D layout -> VGPR r holds row (r) for lanes 0-15, row (r+8) for lanes 16-31.
    const float bv = bias[ncol];
    const int rbase = row0 + ((lane < 16) ? 0 : 8);
#pragma unroll
    for (int r = 0; r < 8; ++r) {
        float v = acc[r] + bv;
        if (relu) v = fmaxf(v, 0.0f);
        out[(long long)(rbase + r) * DFEAT + ncol] = v;
    }
}

// ---------------------------------------------------------------- launcher
extern "C" void kernel_launch(void* const* d_in, const int* in_sizes, int n_in,
                              void* d_out, int out_size, void* d_ws, size_t ws_size,
                              hipStream_t stream) {
    const float* x = (const float*)d_in[0];
    const long long* ei = (const long long*)d_in[1];
    const float* Wl0 = (const float*)d_in[2];
    const float* Wr0 = (const float*)d_in[3];
    const float* b0 = (const float*)d_in[4];
    const float* Wl1 = (const float*)d_in[5];
    const float* Wr1 = (const float*)d_in[6];
    const float* b1 = (const float*)d_in[7];
    const float* Wl2 = (const float*)d_in[8];
    const float* Wr2 = (const float*)d_in[9];
    const float* b2 = (const float*)d_in[10];

    const long long* src = ei;
    const long long* dst = ei + N_EDGES;

    float* cnt = (float*)d_ws;                          // N floats
    float* agg = cnt + N_NODES;                         // N*64 floats (16B-aligned offset)
    float* h1 = agg + (long long)N_NODES * DFEAT;       // N*64 floats
    float* outp = (float*)d_out;                        // N*64 floats (doubles as h2)

    const int featElems = N_NODES * DFEAT;
    dim3 blk256(256), blk128(128);
    int gZeroCnt = (N_NODES / 4 + 255) / 256;
    int gZeroAgg = (featElems / 4 + 255) / 256;
    int gEdges = (N_EDGES + 255) / 256;
    int gScatter = N_EDGES / 8;       // 1.6M % 8 == 0
    int gTiles = N_NODES / 16;        // 100000 % 16 == 0

    // degrees (shared by all layers)
    sage_zero_f32<<<gZeroCnt, blk256, 0, stream>>>(cnt, N_NODES / 4);
    sage_count<<<gEdges, blk256, 0, stream>>>(dst, cnt, N_EDGES);

    // layer 0: x -> h1 (relu)
    sage_zero_f32<<<gZeroAgg, blk256, 0, stream>>>(agg, featElems / 4);
    sage_scatter<<<gScatter, blk256, 0, stream>>>(src, dst, x, agg, N_EDGES);
    sage_gemm<<<gTiles, blk128, 0, stream>>>(agg, cnt, x, Wl0, Wr0, b0, h1, 1);

    // layer 1: h1 -> d_out (relu)
    sage_zero_f32<<<gZeroAgg, blk256, 0, stream>>>(agg, featElems / 4);
    sage_scatter<<<gScatter, blk256, 0, stream>>>(src, dst, h1, agg, N_EDGES);
    sage_gemm<<<gTiles, blk128, 0, stream>>>(agg, cnt, h1, Wl1, Wr1, b1, outp, 1);

    // layer 2: d_out -> d_out (no relu); safe: each block reads its rows before writing
    sage_zero_f32<<<gZeroAgg, blk256, 0, stream>>>(agg, featElems / 4);
    sage_scatter<<<gScatter, blk256, 0, stream>>>(src, dst, outp, agg, N_EDGES);
    sage_gemm<<<gTiles, blk128, 0, stream>>>(agg, cnt, outp, Wl2, Wr2, b2, outp, 0);
}